// InvariantPointAttention_38757784879832
// MI455X (gfx1250) — compile-verified
//
#include <hip/hip_runtime.h>
#include <hip/hip_bf16.h>
#include <math.h>

// ---------------- problem constants ----------------
#define BB   2
#define LL   384
#define DD   128
#define NH   8
#define DHS_ 16
#define DHP_ 4
#define BL   (BB*LL)        // 768
#define BNL  (BB*NH*LL)     // 6144
#define DOUT 1280           // 128 + 1024 + 96 + 32
#define LT   24             // LL/16

typedef float v2f __attribute__((ext_vector_type(2)));
typedef float v8f __attribute__((ext_vector_type(8)));

// D = A(16x4 f32) * B(4x16 f32) + C(16x16 f32), full-precision WMMA.
static __device__ __forceinline__ v8f wmma4(v2f a, v2f b, v8f c) {
  return __builtin_amdgcn_wmma_f32_16x16x4_f32(false, a, false, b, (short)0, c,
                                               false, false);
}

// ---------------------------------------------------------------------------
// Generic row-major WMMA GEMM: C[M,Nc] = A[M,K] @ W[K,Nc] (+ bias[Nc]).
// One wave per 16x16 tile; tile id is wave-uniform (EXEC all-ones at WMMA).
// ---------------------------------------------------------------------------
__global__ void wmma_gemm_kernel(const float* __restrict__ A,
                                 const float* __restrict__ W,
                                 const float* __restrict__ bias,
                                 float* __restrict__ C,
                                 int M, int K, int Nc, int total_tiles) {
  int tile = blockIdx.x * 8 + (threadIdx.x >> 5);
  if (tile >= total_tiles) return;
  int lane = threadIdx.x & 31;
  int h = lane >> 4, l16 = lane & 15;
  int ntn = Nc >> 4;
  int tm = (tile / ntn) << 4;
  int tn = (tile % ntn) << 4;
  const float* ap = A + (size_t)(tm + l16) * K + 2 * h;
  const float* wp = W + (size_t)(2 * h) * Nc + tn + l16;
  v8f acc = {};
  for (int k = 0; k < K; k += 4) {
    v2f a, b;
    a.x = ap[k];
    a.y = ap[k + 1];
    b.x = wp[(size_t)k * Nc];
    b.y = wp[(size_t)(k + 1) * Nc];
    acc = wmma4(a, b, acc);
  }
  float bv = bias ? bias[tn + l16] : 0.0f;
#pragma unroll
  for (int v = 0; v < 8; ++v) {
    int row = tm + v + (h << 3);
    C[(size_t)row * Nc + tn + l16] = acc[v] + bv;
  }
}

// ---------------------------------------------------------------------------
// bias_pair[b,n,i,j] = sum_d e[b,i,j,d] * Wb[d,n].  N=8 is too narrow for a
// WMMA tile to win; stream e once, coalesced, through LDS.
// ---------------------------------------------------------------------------
__global__ void bias_kernel(const float* __restrict__ e,
                            const float* __restrict__ Wb,
                            float* __restrict__ bias) {
  __shared__ float se[32 * 129];   // 32 j-rows x 128 d, padded vs bank conflicts
  __shared__ float swb[128 * 8];
  int blk = blockIdx.x;            // b*(LL*12) + i*12 + jb
  int jb = blk % 12;
  int i  = (blk / 12) % LL;
  int b  = blk / (12 * LL);
  const float* ep = e + (((size_t)b * LL + i) * LL + (size_t)jb * 32) * DD;
  for (int idx = threadIdx.x; idx < 32 * 128; idx += 256) {
    int jj = idx >> 7, d = idx & 127;
    se[jj * 129 + d] = ep[idx];
  }
  for (int idx = threadIdx.x; idx < 128 * 8; idx += 256) swb[idx] = Wb[idx];
  __syncthreads();
  int n = threadIdx.x >> 5, jj = threadIdx.x & 31;
  float s = 0.0f;
  for (int d = 0; d < 128; ++d) s += se[jj * 129 + d] * swb[d * 8 + n];
  bias[(((size_t)b * NH + n) * LL + i) * LL + jb * 32 + jj] = s;
}

// ---------------------------------------------------------------------------
// qp[b,n,l,p,c] = sum_k hp[b,n,l,p,k] * r[b,l,k,c] + t[b,l,c];  optional sumsq.
// ---------------------------------------------------------------------------
__global__ void point_transform_kernel(const float* __restrict__ hmat,
                                       const float* __restrict__ r,
                                       const float* __restrict__ t,
                                       float* __restrict__ outp,
                                       float* __restrict__ sumsq) {
  int idx = blockIdx.x * blockDim.x + threadIdx.x;  // (b*NH+n)*LL + l
  if (idx >= BNL) return;
  int l = idx % LL;
  int n = (idx / LL) % NH;
  int b = idx / (NH * LL);
  const float* hp = hmat + (size_t)(b * LL + l) * (NH * 12) + n * 12;
  const float* rp = r + (size_t)(b * LL + l) * 9;
  const float* tp = t + (size_t)(b * LL + l) * 3;
  float* op = outp + (size_t)idx * 12;
  float ss = 0.0f;
#pragma unroll
  for (int p = 0; p < 4; ++p)
#pragma unroll
    for (int c = 0; c < 3; ++c) {
      float v = tp[c];
#pragma unroll
      for (int k = 0; k < 3; ++k) v += hp[p * 3 + k] * rp[k * 3 + c];
      op[p * 3 + c] = v;
      ss += v * v;
    }
  if (sumsq) sumsq[idx] = ss;
}

// ---------------------------------------------------------------------------
// logit[b,n,i,j] = 3^-.5 * ( .25 * qs.ks + bias + (-.125*gamma)*dist2 )
// Two WMMA passes per tile: K=16 (scalar) and K=12 (points).
// ---------------------------------------------------------------------------
__global__ void logit_kernel(const float* __restrict__ qs,
                             const float* __restrict__ ks,
                             const float* __restrict__ qp,
                             const float* __restrict__ kp,
                             const float* __restrict__ qq,
                             const float* __restrict__ kk,
                             const float* __restrict__ bias,
                             const float* __restrict__ gamma,
                             float* __restrict__ logit) {
  int tile = blockIdx.x * 8 + (threadIdx.x >> 5);  // BB*NH*LT*LT = 9216
  int jt = tile % LT;
  int it = (tile / LT) % LT;
  int n  = (tile / (LT * LT)) % NH;
  int b  = tile / (LT * LT * NH);
  int lane = threadIdx.x & 31;
  int h = lane >> 4, l16 = lane & 15;

  const float* qsp = qs + (size_t)(b * LL + it * 16 + l16) * DD + n * DHS_ + 2 * h;
  const float* ksp = ks + (size_t)(b * LL + jt * 16 + l16) * DD + n * DHS_ + 2 * h;
  v8f accS = {};
#pragma unroll
  for (int k = 0; k < 16; k += 4) {
    v2f a, bv2;
    a.x = qsp[k];   a.y = qsp[k + 1];
    bv2.x = ksp[k]; bv2.y = ksp[k + 1];
    accS = wmma4(a, bv2, accS);
  }
  const float* qpp = qp + (size_t)((b * NH + n) * LL + it * 16 + l16) * 12 + 2 * h;
  const float* kpp = kp + (size_t)((b * NH + n) * LL + jt * 16 + l16) * 12 + 2 * h;
  v8f accP = {};
#pragma unroll
  for (int k = 0; k < 12; k += 4) {
    v2f a, bv2;
    a.x = qpp[k];   a.y = qpp[k + 1];
    bv2.x = kpp[k]; bv2.y = kpp[k + 1];
    accP = wmma4(a, bv2, accP);
  }
  size_t hb = (size_t)(b * NH + n) * LL;
  int j = jt * 16 + l16;
  float kkj = kk[hb + j];
  float g = gamma[n];
  const float rsqrt3 = 0.57735026918962576f;
#pragma unroll
  for (int v = 0; v < 8; ++v) {
    int i = it * 16 + v + (h << 3);
    float d2 = qq[hb + i] + kkj - 2.0f * accP[v];
    float lp = -0.125f * g * d2;
    float bsv = bias[(hb + i) * LL + j];
    logit[(hb + i) * LL + j] = rsqrt3 * (0.25f * accS[v] + bsv + lp);
  }
}

// ---------------------------------------------------------------------------
// Row softmax over j (384) — one wave per row, wave32 shuffle reductions.
// ---------------------------------------------------------------------------
__global__ void softmax_kernel(float* __restrict__ attn) {
  int row = blockIdx.x * 8 + (threadIdx.x >> 5);  // BNL rows
  int lane = threadIdx.x & 31;
  float* p = attn + (size_t)row * LL;
  float v[12];
  float m = -1e30f;
#pragma unroll
  for (int i = 0; i < 12; ++i) {
    v[i] = p[lane + i * 32];
    m = fmaxf(m, v[i]);
  }
#pragma unroll
  for (int off = 16; off > 0; off >>= 1) m = fmaxf(m, __shfl_xor(m, off, 32));
  float s = 0.0f;
#pragma unroll
  for (int i = 0; i < 12; ++i) {
    v[i] = __expf(v[i] - m);
    s += v[i];
  }
#pragma unroll
  for (int off = 16; off > 0; off >>= 1) s += __shfl_xor(s, off, 32);
  float inv = 1.0f / s;
#pragma unroll
  for (int i = 0; i < 12; ++i) p[lane + i * 32] = v[i] * inv;
}

// ---------------------------------------------------------------------------
// out_scalar = attn @ vs (16 cols) and op = attn @ vp (12 cols, padded to 16);
// shared A-fragments, two accumulators.  K = 384.
// ---------------------------------------------------------------------------
__global__ void attn_v_kernel(const float* __restrict__ attn,
                              const float* __restrict__ vs,
                              const float* __restrict__ vp,
                              float* __restrict__ out_cat,
                              float* __restrict__ op) {
  int tile = blockIdx.x * 8 + (threadIdx.x >> 5);  // BB*NH*LT = 384
  int it = tile % LT;
  int n  = (tile / LT) % NH;
  int b  = tile / (LT * NH);
  int lane = threadIdx.x & 31;
  int h = lane >> 4, l16 = lane & 15;
  size_t hb = (size_t)(b * NH + n) * LL;
  const float* ap  = attn + (hb + it * 16 + l16) * LL + 2 * h;
  const float* vsb = vs + (size_t)b * LL * DD + n * DHS_ + l16;
  const float* vpb = vp + hb * 12 + l16;
  bool pv = l16 < 12;
  v8f accS = {}, accP = {};
  for (int k = 0; k < LL; k += 4) {
    v2f a;
    a.x = ap[k];
    a.y = ap[k + 1];
    v2f bs;
    bs.x = vsb[(size_t)(k + 2 * h) * DD];
    bs.y = vsb[(size_t)(k + 2 * h + 1) * DD];
    accS = wmma4(a, bs, accS);
    v2f bp;
    bp.x = pv ? vpb[(k + 2 * h) * 12] : 0.0f;
    bp.y = pv ? vpb[(k + 2 * h + 1) * 12] : 0.0f;
    accP = wmma4(a, bp, accP);
  }
#pragma unroll
  for (int v = 0; v < 8; ++v) {
    int i = it * 16 + v + (h << 3);
    out_cat[(size_t)(b * LL + i) * DOUT + n * DHS_ + l16] = accS[v];
    if (pv) op[(hb + i) * 12 + l16] = accP[v];
  }
}

// ---------------------------------------------------------------------------
// out_pair[b,n,i,d] = sum_j attn[b,n,i,j] * e[b,i,j,d].  Per (b,i,d-tile):
// A is 8 valid head-rows padded to 16, B streams e rows (with prefetch).
// ---------------------------------------------------------------------------
__global__ void out_pair_kernel(const float* __restrict__ attn,
                                const float* __restrict__ e,
                                float* __restrict__ out_cat) {
  int tile = blockIdx.x * 8 + (threadIdx.x >> 5);  // BB*LL*8 = 6144
  int dt = tile % 8;
  int i  = (tile / 8) % LL;
  int b  = tile / (8 * LL);
  int lane = threadIdx.x & 31;
  int h = lane >> 4, l16 = lane & 15;
  bool av = l16 < 8;  // rows 8..15 of A are zero padding
  const float* ap = attn + ((size_t)(b * NH + l16) * LL + i) * LL + 2 * h;
  const float* ep = e + ((size_t)(b * LL + i) * LL) * DD + dt * 16 + l16;
  v8f acc = {};
  for (int k = 0; k < LL; k += 4) {
    if ((k & 63) == 0 && (k + 64) < LL)
      __builtin_prefetch(ep + (size_t)(k + 64) * DD, 0, 0);  // global_prefetch_b8
    v2f a;
    a.x = av ? ap[k] : 0.0f;
    a.y = av ? ap[k + 1] : 0.0f;
    v2f bv;
    bv.x = ep[(size_t)(k + 2 * h) * DD];
    bv.y = ep[(size_t)(k + 2 * h + 1) * DD];
    acc = wmma4(a, bv, acc);
  }
  if (h == 0) {  // only M=0..7 rows (heads) are valid
#pragma unroll
    for (int v = 0; v < 8; ++v)
      out_cat[(size_t)(b * LL + i) * DOUT + DD + v * DD + dt * 16 + l16] = acc[v];
  }
}

// ---------------------------------------------------------------------------
// op_local = (op - t) rotated by r^T; writes out_point and out_norm slices.
// ---------------------------------------------------------------------------
__global__ void op_finish_kernel(const float* __restrict__ op,
                                 const float* __restrict__ r,
                                 const float* __restrict__ t,
                                 float* __restrict__ out_cat) {
  int idx = blockIdx.x * blockDim.x + threadIdx.x;  // (b*NH+n)*LL + i
  if (idx >= BNL) return;
  int i = idx % LL;
  int n = (idx / LL) % NH;
  int b = idx / (NH * LL);
  const float* opv = op + (size_t)idx * 12;
  const float* rp = r + (size_t)(b * LL + i) * 9;
  const float* tp = t + (size_t)(b * LL + i) * 3;
  float* oc = out_cat + (size_t)(b * LL + i) * DOUT;
#pragma unroll
  for (int p = 0; p < 4; ++p) {
    float om[3];
#pragma unroll
    for (int k = 0; k < 3; ++k) om[k] = opv[p * 3 + k] - tp[k];
    float nsq = 0.0f;
#pragma unroll
    for (int c = 0; c < 3; ++c) {
      float v = om[0] * rp[c * 3 + 0] + om[1] * rp[c * 3 + 1] + om[2] * rp[c * 3 + 2];
      oc[DD + NH * DD + n * 12 + p * 3 + c] = v;  // out_point @ 1152
      nsq += v * v;
    }
    oc[DD + NH * DD + NH * 12 + n * 4 + p] = sqrtf(nsq);  // out_norm @ 1248
  }
}

// ---------------------------------------------------------------------------
extern "C" void kernel_launch(void* const* d_in, const int* in_sizes, int n_in,
                              void* d_out, int out_size, void* d_ws, size_t ws_size,
                              hipStream_t stream) {
  (void)in_sizes; (void)n_in; (void)out_size;
  const float* x     = (const float*)d_in[0];
  const float* e     = (const float*)d_in[1];
  const float* r     = (const float*)d_in[2];
  const float* t     = (const float*)d_in[3];
  const float* Wq_s  = (const float*)d_in[4];
  const float* Wk_s  = (const float*)d_in[5];
  const float* Wv_s  = (const float*)d_in[6];
  const float* Wb    = (const float*)d_in[7];
  const float* Wq_p  = (const float*)d_in[8];
  const float* Wk_p  = (const float*)d_in[9];
  const float* Wv_p  = (const float*)d_in[10];
  const float* gamma = (const float*)d_in[11];
  const float* Wo    = (const float*)d_in[12];
  const float* bo    = (const float*)d_in[13];
  float* out = (float*)d_out;
  float* ws  = (float*)d_ws;

  // ---- workspace layout (floats) ----
  float* qs   = ws;                 // 768*128
  float* ks_  = qs  + 98304;
  float* vs   = ks_ + 98304;
  float* hq   = vs  + 98304;        // 768*96
  float* hk   = hq  + 73728;
  float* hv   = hk  + 73728;
  float* qp   = hv  + 73728;        // BNL*12
  float* kp   = qp  + 73728;
  float* vp   = kp  + 73728;
  float* qq   = vp  + 73728;        // BNL
  float* kk   = qq  + 6144;
  float* bias = kk  + 6144;         // BNL*LL
  float* attn = bias + 2359296;     // BNL*LL
  float* op   = attn + 2359296;     // BNL*12
  float* ocat = op  + 73728;        // 768*1280
  size_t req = (size_t)((ocat + 983040) - ws) * sizeof(float);
  if (ws_size < req) return;

  dim3 blk(256);
  auto gemm = [&](const float* A, const float* W, const float* bvec, float* C,
                  int M, int K, int Nc) {
    int tiles = (M / 16) * (Nc / 16);
    wmma_gemm_kernel<<<(tiles + 7) / 8, blk, 0, stream>>>(A, W, bvec, C, M, K,
                                                          Nc, tiles);
  };

  // projections (f32 WMMA)
  gemm(x, Wq_s, nullptr, qs, BL, DD, DD);
  gemm(x, Wk_s, nullptr, ks_, BL, DD, DD);
  gemm(x, Wv_s, nullptr, vs, BL, DD, DD);
  gemm(x, Wq_p, nullptr, hq, BL, DD, NH * 12);
  gemm(x, Wk_p, nullptr, hk, BL, DD, NH * 12);
  gemm(x, Wv_p, nullptr, hv, BL, DD, NH * 12);

  // pair bias (streams e once, coalesced through LDS)
  bias_kernel<<<BB * LL * 12, blk, 0, stream>>>(e, Wb, bias);

  // frame transforms + squared norms
  point_transform_kernel<<<BNL / 256, blk, 0, stream>>>(hq, r, t, qp, qq);
  point_transform_kernel<<<BNL / 256, blk, 0, stream>>>(hk, r, t, kp, kk);
  point_transform_kernel<<<BNL / 256, blk, 0, stream>>>(hv, r, t, vp, nullptr);

  // logits (two WMMA passes) + softmax
  logit_kernel<<<(BB * NH * LT * LT) / 8, blk, 0, stream>>>(qs, ks_, qp, kp, qq,
                                                            kk, bias, gamma, attn);
  softmax_kernel<<<BNL / 8, blk, 0, stream>>>(attn);

  // attention-weighted reductions
  attn_v_kernel<<<(BB * NH * LT) / 8, blk, 0, stream>>>(attn, vs, vp, ocat, op);
  out_pair_kernel<<<(BB * LL * 8) / 8, blk, 0, stream>>>(attn, e, ocat);
  op_finish_kernel<<<BNL / 256, blk, 0, stream>>>(op, r, t, ocat);

  // final projection: out = ocat @ Wo + bo
  gemm(ocat, Wo, bo, out, BL, DOUT, DD);
}